// RibSegModel_39908836114823
// MI455X (gfx1250) — compile-verified
//
#include <hip/hip_runtime.h>
#include <math.h>

typedef __attribute__((ext_vector_type(2))) float v2f;
typedef __attribute__((ext_vector_type(8))) float v8f;

#define BN_SCALE 0.99999500003749968f

// ---------------------------------------------------------------------------
// transpose (B,C,N) -> (B,N,C)
// ---------------------------------------------------------------------------
__global__ void k_transpose(const float* __restrict__ in, float* __restrict__ out,
                            int B, int C, int N) {
  size_t total = (size_t)B * C * N;
  for (size_t i = (size_t)blockIdx.x * blockDim.x + threadIdx.x; i < total;
       i += (size_t)gridDim.x * blockDim.x) {
    int n = (int)(i % N);
    int c = (int)((i / N) % C);
    int b = (int)(i / ((size_t)N * C));
    out[((size_t)b * N + n) * C + c] = in[i];
  }
}

// ---------------------------------------------------------------------------
// Farthest point sampling: one block (1024 thr) per batch, dists in registers,
// LDS argmax tree. Emits `far` BEFORE updating (matches jax.lax.scan body).
// ---------------------------------------------------------------------------
#define FPS_T 1024
#define FPS_MAXP 16
__global__ void k_fps(const float* __restrict__ pts, int* __restrict__ out_idx,
                      int N, int npoint) {
  __shared__ float s_val[FPS_T];
  __shared__ int s_idx[FPS_T];
  int b = blockIdx.x;
  int tid = threadIdx.x;
  const float* P = pts + (size_t)b * N * 3;
  float dist[FPS_MAXP];
  for (int t = 0; t < FPS_MAXP; ++t) dist[t] = 1e10f;
  int far = 0;
  for (int it = 0; it < npoint; ++it) {
    if (tid == 0) out_idx[b * npoint + it] = far;
    float cx = P[far * 3 + 0], cy = P[far * 3 + 1], cz = P[far * 3 + 2];
    float best = -1.0f;
    int besti = 0;
    for (int t = 0; t < FPS_MAXP; ++t) {
      int p = t * FPS_T + tid;
      if (p >= N) break;
      float dx = P[p * 3 + 0] - cx;
      float dy = P[p * 3 + 1] - cy;
      float dz = P[p * 3 + 2] - cz;
      float d = dx * dx + dy * dy + dz * dz;
      float dm = dist[t];
      dm = d < dm ? d : dm;
      dist[t] = dm;
      if (dm > best) { best = dm; besti = p; }   // strict > keeps lowest index
    }
    s_val[tid] = best;
    s_idx[tid] = besti;
    __syncthreads();
    for (int s = FPS_T / 2; s > 0; s >>= 1) {
      if (tid < s) {
        float v2 = s_val[tid + s];
        int i2 = s_idx[tid + s];
        if (v2 > s_val[tid] || (v2 == s_val[tid] && i2 < s_idx[tid])) {
          s_val[tid] = v2; s_idx[tid] = i2;
        }
      }
      __syncthreads();
    }
    far = s_idx[0];
    __syncthreads();
  }
}

// ---------------------------------------------------------------------------
// gather: out[b,s,:] = in[b, idx[b,s], :]
// ---------------------------------------------------------------------------
__global__ void k_gather(const float* __restrict__ in, const int* __restrict__ idx,
                         float* __restrict__ out, int B, int N, int S, int C) {
  size_t total = (size_t)B * S * C;
  for (size_t i = (size_t)blockIdx.x * blockDim.x + threadIdx.x; i < total;
       i += (size_t)gridDim.x * blockDim.x) {
    int c = (int)(i % C);
    int s = (int)((i / C) % S);
    int b = (int)(i / ((size_t)S * C));
    int id = idx[b * S + s];
    if (id > N - 1) id = N - 1;
    if (id < 0) id = 0;
    out[i] = in[((size_t)b * N + id) * C + c];
  }
}

// ---------------------------------------------------------------------------
// ball query: one wave32 per center; ordered compaction via ballot.
// In-range test mirrors reference: !(d > r^2), d in expanded |a|^2+|b|^2-2ab.
// ---------------------------------------------------------------------------
__global__ void k_query_ball(const float* __restrict__ xyz,
                             const float* __restrict__ new_xyz,
                             int* __restrict__ gidx,
                             int B, int N, int S, int k, float r2) {
  int wave = (int)(((size_t)blockIdx.x * blockDim.x + threadIdx.x) >> 5);
  int lane = threadIdx.x & 31;
  if (wave >= B * S) return;
  int b = wave / S, s = wave % S;
  float cx = new_xyz[((size_t)b * S + s) * 3 + 0];
  float cy = new_xyz[((size_t)b * S + s) * 3 + 1];
  float cz = new_xyz[((size_t)b * S + s) * 3 + 2];
  float c2 = cx * cx + cy * cy + cz * cz;
  int* out = gidx + ((size_t)b * S + s) * k;
  int count = 0;
  int first = N - 1;
  for (int base = 0; base < N && count < k; base += 32) {
    int p = base + lane;
    bool in = false;
    if (p < N) {
      float px = xyz[((size_t)b * N + p) * 3 + 0];
      float py = xyz[((size_t)b * N + p) * 3 + 1];
      float pz = xyz[((size_t)b * N + p) * 3 + 2];
      float d = c2 + (px * px + py * py + pz * pz) - 2.0f * (cx * px + cy * py + cz * pz);
      in = !(d > r2);
    }
    unsigned int mask = (unsigned int)__ballot(in);   // wave32: low 32 bits
    while (mask && count < k) {
      int bit = __ffs((int)mask) - 1;
      mask &= mask - 1u;
      int id = base + bit;
      if (count == 0) first = id;
      if (lane == 0) out[count] = id;
      ++count;
    }
  }
  for (int j = count + lane; j < k; j += 32) out[j] = first;
}

// ---------------------------------------------------------------------------
// build grouped features: out[row, :] = [ points[b,gidx,:Cp], xyz[b,gidx]-center ]
// rows = B*S*k, C = Cp+3
// ---------------------------------------------------------------------------
__global__ void k_group_concat(const float* __restrict__ xyz,
                               const float* __restrict__ points,
                               const float* __restrict__ new_xyz,
                               const int* __restrict__ gidx,
                               float* __restrict__ out,
                               int B, int N, int S, int k, int Cp) {
  int C = Cp + 3;
  size_t total = (size_t)B * S * k * C;
  for (size_t i = (size_t)blockIdx.x * blockDim.x + threadIdx.x; i < total;
       i += (size_t)gridDim.x * blockDim.x) {
    int c = (int)(i % C);
    size_t row = i / C;
    int j = (int)(row % k);
    size_t bs = row / k;
    int s = (int)(bs % S);
    int b = (int)(bs / S);
    int id = gidx[bs * k + j];
    if (id > N - 1) id = N - 1;
    if (id < 0) id = 0;
    float v;
    if (c < Cp) {
      v = points[((size_t)b * N + id) * Cp + c];
    } else {
      int d = c - Cp;
      v = xyz[((size_t)b * N + id) * 3 + d] - new_xyz[((size_t)b * S + s) * 3 + d];
    }
    out[i] = v;
  }
}

// ---------------------------------------------------------------------------
// concat2: out[row,:] = [A[row,:C1], Bm[row,:C2]]
// ---------------------------------------------------------------------------
__global__ void k_concat2(const float* __restrict__ A, int C1,
                          const float* __restrict__ Bm, int C2,
                          float* __restrict__ out, size_t rows) {
  int C = C1 + C2;
  size_t total = rows * C;
  for (size_t i = (size_t)blockIdx.x * blockDim.x + threadIdx.x; i < total;
       i += (size_t)gridDim.x * blockDim.x) {
    int c = (int)(i % C);
    size_t r = i / C;
    out[i] = (c < C1) ? A[r * C1 + c] : Bm[r * C2 + (c - C1)];
  }
}

// ---------------------------------------------------------------------------
// fused GEMM + bias + activation on fp32 WMMA (V_WMMA_F32_16X16X4_F32).
// One wave32 owns a 16x64 output strip: 4 accumulators reuse one A fragment,
// 4 back-to-back WMMAs per K-step. Hot loop properties:
//  - no conditional loads (row/col addresses clamped; OOB lanes only pollute
//    outputs that are masked at store time)
//  - running pointers only: A pointer advances 16 B/step, each B-column
//    pointer advances 4*Cout*4 B/step; the odd W row is reached through the
//    load's immediate offset (Cout*4 <= 4KB fits the 24-bit IOFFSET)
//  - K tail (Cin % 4) handled once after the loop with value-selects
//   mode 0: y = x*W + b
//   mode 1: y = relu((x*W + b) * BN_SCALE)
//   mode 2: y = sigmoid(x*W + b)
// ---------------------------------------------------------------------------
__global__ void __launch_bounds__(256)
k_gemm_wmma(const float* __restrict__ X, const float* __restrict__ W,
            const float* __restrict__ bias, float* __restrict__ Y,
            int rows, int Cin, int Cout, int mode) {
  int lane = threadIdx.x & 31;
  int wib = threadIdx.x >> 5;
  int tile = blockIdx.x * (blockDim.x >> 5) + wib;
  int cs = (Cout + 63) >> 6;              // 64-wide column strips
  int rt = (rows + 15) >> 4;
  if (tile >= rt * cs) return;            // wave-uniform exit
  int rowTile = tile / cs;
  int colStrip = tile - rowTile * cs;
  int half = lane >> 4;                   // K sub-pair select
  int l15 = lane & 15;
  int row0 = rowTile << 4;
  int col0 = colStrip << 6;

  int r = row0 + l15;
  int rc = r < rows ? r : rows - 1;       // clamped row (stores masked later)
  const float* xrow = X + (size_t)rc * Cin;

  int cidx[4];
  int cclamp[4];
#pragma unroll
  for (int t = 0; t < 4; ++t) {
    int c = col0 + (t << 4) + l15;
    cidx[t] = c;
    cclamp[t] = c < Cout ? c : Cout - 1;  // clamped col (stores masked later)
  }

  v8f acc0 = {}, acc1 = {}, acc2 = {}, acc3 = {};
  int kmain = Cin & ~3;
  size_t wstep = (size_t)Cout * 4;        // 4 K-rows per step

  // running pointers (no per-iteration 64-bit multiplies)
  const float* pa = xrow + (half << 1);
  const float* pw0 = W + (size_t)(half << 1) * Cout + cclamp[0];
  const float* pw1 = W + (size_t)(half << 1) * Cout + cclamp[1];
  const float* pw2 = W + (size_t)(half << 1) * Cout + cclamp[2];
  const float* pw3 = W + (size_t)(half << 1) * Cout + cclamp[3];

  for (int k0 = 0; k0 < kmain; k0 += 4) {
    v2f a;
    a.x = pa[0];
    a.y = pa[1];
    v2f b0, b1, b2, b3;
    b0.x = pw0[0]; b0.y = pw0[Cout];      // second row via immediate offset
    b1.x = pw1[0]; b1.y = pw1[Cout];
    b2.x = pw2[0]; b2.y = pw2[Cout];
    b3.x = pw3[0]; b3.y = pw3[Cout];
    acc0 = __builtin_amdgcn_wmma_f32_16x16x4_f32(false, a, false, b0, (short)0, acc0, false, false);
    acc1 = __builtin_amdgcn_wmma_f32_16x16x4_f32(false, a, false, b1, (short)0, acc1, false, false);
    acc2 = __builtin_amdgcn_wmma_f32_16x16x4_f32(false, a, false, b2, (short)0, acc2, false, false);
    acc3 = __builtin_amdgcn_wmma_f32_16x16x4_f32(false, a, false, b3, (short)0, acc3, false, false);
    pa += 4;
    pw0 += wstep; pw1 += wstep; pw2 += wstep; pw3 += wstep;
  }

  if (kmain < Cin) {                      // K tail: zero A via value selects
    int ka = kmain + (half << 1);
    int ka0 = ka < Cin ? ka : Cin - 1;
    int ka1 = (ka + 1) < Cin ? (ka + 1) : Cin - 1;
    v2f a;
    a.x = ka < Cin ? xrow[ka0] : 0.0f;
    a.y = (ka + 1) < Cin ? xrow[ka1] : 0.0f;
    const float* w0 = W + (size_t)ka0 * Cout;
    const float* w1 = W + (size_t)ka1 * Cout;
    v2f b0, b1, b2, b3;
    b0.x = w0[cclamp[0]]; b0.y = w1[cclamp[0]];
    b1.x = w0[cclamp[1]]; b1.y = w1[cclamp[1]];
    b2.x = w0[cclamp[2]]; b2.y = w1[cclamp[2]];
    b3.x = w0[cclamp[3]]; b3.y = w1[cclamp[3]];
    acc0 = __builtin_amdgcn_wmma_f32_16x16x4_f32(false, a, false, b0, (short)0, acc0, false, false);
    acc1 = __builtin_amdgcn_wmma_f32_16x16x4_f32(false, a, false, b1, (short)0, acc1, false, false);
    acc2 = __builtin_amdgcn_wmma_f32_16x16x4_f32(false, a, false, b2, (short)0, acc2, false, false);
    acc3 = __builtin_amdgcn_wmma_f32_16x16x4_f32(false, a, false, b3, (short)0, acc3, false, false);
  }

  // epilogue: bias + activation + masked stores (straight-line, no indexing
  // through a local array so accumulators never touch scratch)
#define EPILOGUE(ACC, T)                                                      \
  if (cidx[T] < Cout) {                                                       \
    float bv = bias[cidx[T]];                                                 \
    _Pragma("unroll")                                                         \
    for (int q = 0; q < 8; ++q) {                                             \
      int rr = row0 + q + (half << 3);   /* C/D layout: M = q + 8*half */     \
      if (rr >= rows) continue;                                               \
      float v = ACC[q] + bv;                                                  \
      if (mode == 1) {                                                        \
        v *= BN_SCALE;                                                        \
        v = v > 0.0f ? v : 0.0f;                                              \
      } else if (mode == 2) {                                                 \
        v = 1.0f / (1.0f + expf(-v));                                         \
      }                                                                       \
      Y[(size_t)rr * Cout + cidx[T]] = v;                                     \
    }                                                                         \
  }
  EPILOGUE(acc0, 0)
  EPILOGUE(acc1, 1)
  EPILOGUE(acc2, 2)
  EPILOGUE(acc3, 3)
#undef EPILOGUE
}

// ---------------------------------------------------------------------------
// max-pool over k: in rows = G*k, C.  out[g*out_ld + out_off + c] = max_j in[...]
// ---------------------------------------------------------------------------
__global__ void k_maxpool(const float* __restrict__ in, float* __restrict__ out,
                          int G, int k, int C, int out_ld, int out_off) {
  size_t total = (size_t)G * C;
  for (size_t i = (size_t)blockIdx.x * blockDim.x + threadIdx.x; i < total;
       i += (size_t)gridDim.x * blockDim.x) {
    int c = (int)(i % C);
    int g = (int)(i / C);
    float m = -3.4e38f;
    const float* p = in + ((size_t)g * k) * C + c;
    for (int j = 0; j < k; ++j) {
      float v = p[(size_t)j * C];
      m = v > m ? v : m;
    }
    out[(size_t)g * out_ld + out_off + c] = m;
  }
}

// ---------------------------------------------------------------------------
// 3-NN (expanded sqdist, stable tie-break like top_k) + inverse-dist weights
// ---------------------------------------------------------------------------
__global__ void k_nn3(const float* __restrict__ xyz1, const float* __restrict__ xyz2,
                      int* __restrict__ idx3, float* __restrict__ w3,
                      int B, int Nq, int M) {
  size_t total = (size_t)B * Nq;
  for (size_t i = (size_t)blockIdx.x * blockDim.x + threadIdx.x; i < total;
       i += (size_t)gridDim.x * blockDim.x) {
    int n = (int)(i % Nq);
    int b = (int)(i / Nq);
    float ax = xyz1[((size_t)b * Nq + n) * 3 + 0];
    float ay = xyz1[((size_t)b * Nq + n) * 3 + 1];
    float az = xyz1[((size_t)b * Nq + n) * 3 + 2];
    float a2 = ax * ax + ay * ay + az * az;
    float d0 = 3.4e38f, d1 = 3.4e38f, d2 = 3.4e38f;
    int i0 = 0, i1 = 0, i2 = 0;
    for (int m = 0; m < M; ++m) {
      float bx = xyz2[((size_t)b * M + m) * 3 + 0];
      float by = xyz2[((size_t)b * M + m) * 3 + 1];
      float bz = xyz2[((size_t)b * M + m) * 3 + 2];
      float d = a2 + (bx * bx + by * by + bz * bz) -
                2.0f * (ax * bx + ay * by + az * bz);
      if (d < d0)      { d2 = d1; i2 = i1; d1 = d0; i1 = i0; d0 = d; i0 = m; }
      else if (d < d1) { d2 = d1; i2 = i1; d1 = d; i1 = m; }
      else if (d < d2) { d2 = d; i2 = m; }
    }
    float w0 = 1.0f / (d0 + 1e-8f);
    float w1 = 1.0f / (d1 + 1e-8f);
    float w2 = 1.0f / (d2 + 1e-8f);
    float ws = w0 + w1 + w2;
    idx3[i * 3 + 0] = i0; idx3[i * 3 + 1] = i1; idx3[i * 3 + 2] = i2;
    w3[i * 3 + 0] = w0 / ws; w3[i * 3 + 1] = w1 / ws; w3[i * 3 + 2] = w2 / ws;
  }
}

__global__ void k_nn_broadcast(int* __restrict__ idx3, float* __restrict__ w3,
                               size_t n) {
  for (size_t i = (size_t)blockIdx.x * blockDim.x + threadIdx.x; i < n;
       i += (size_t)gridDim.x * blockDim.x) {
    idx3[i * 3 + 0] = 0; idx3[i * 3 + 1] = 0; idx3[i * 3 + 2] = 0;
    w3[i * 3 + 0] = 1.0f; w3[i * 3 + 1] = 0.0f; w3[i * 3 + 2] = 0.0f;
  }
}

// ---------------------------------------------------------------------------
// out[b,n,:] = [p1a[b,n,:C1a], p1b[b,n,:C1b], sum_j w_j * pts2[b,idx_j,:C2]]
// ---------------------------------------------------------------------------
__global__ void k_interp_concat(const float* __restrict__ p1a, int C1a,
                                const float* __restrict__ p1b, int C1b,
                                const float* __restrict__ pts2, int C2,
                                const int* __restrict__ idx3,
                                const float* __restrict__ w3,
                                float* __restrict__ out, int B, int Nq, int M) {
  int C = C1a + C1b + C2;
  size_t total = (size_t)B * Nq * C;
  for (size_t i = (size_t)blockIdx.x * blockDim.x + threadIdx.x; i < total;
       i += (size_t)gridDim.x * blockDim.x) {
    int c = (int)(i % C);
    size_t bn = i / C;
    int b = (int)(bn / Nq);
    float v;
    if (c < C1a) {
      v = p1a[bn * C1a + c];
    } else if (c < C1a + C1b) {
      v = p1b[bn * C1b + (c - C1a)];
    } else {
      int c2 = c - C1a - C1b;
      v = 0.0f;
      for (int j = 0; j < 3; ++j) {
        int id = idx3[bn * 3 + j];
        if (id > M - 1) id = M - 1;
        if (id < 0) id = 0;
        v += w3[bn * 3 + j] * pts2[((size_t)b * M + id) * C2 + c2];
      }
    }
    out[i] = v;
  }
}

__global__ void k_mul(const float* __restrict__ x, const float* __restrict__ a,
                      float* __restrict__ out, size_t n) {
  for (size_t i = (size_t)blockIdx.x * blockDim.x + threadIdx.x; i < n;
       i += (size_t)gridDim.x * blockDim.x)
    out[i] = x[i] * a[i];
}

// ===========================================================================
// Host orchestration
// ===========================================================================
struct Layer { const float* W; const float* b; int cin; int cout; };

extern "C" void kernel_launch(void* const* d_in, const int* in_sizes, int n_in,
                              void* d_out, int out_size, void* d_ws, size_t ws_size,
                              hipStream_t stream) {
  (void)in_sizes; (void)n_in; (void)out_size; (void)ws_size;
  const int B = 8, N = 16384, S1 = 512, S2 = 128;
  const float* xyz_bcn = (const float*)d_in[0];
  auto d = [&](int i) { return (const float*)d_in[i]; };

  Layer sa1m[3][3] = {
    {{d(1), d(2), 6, 32}, {d(3), d(4), 32, 32}, {d(5), d(6), 32, 64}},
    {{d(7), d(8), 6, 64}, {d(9), d(10), 64, 64}, {d(11), d(12), 64, 128}},
    {{d(13), d(14), 6, 64}, {d(15), d(16), 64, 96}, {d(17), d(18), 96, 128}}};
  Layer sa2m[2][3] = {
    {{d(19), d(20), 323, 128}, {d(21), d(22), 128, 128}, {d(23), d(24), 128, 256}},
    {{d(25), d(26), 323, 128}, {d(27), d(28), 128, 196}, {d(29), d(30), 196, 256}}};
  Layer sa3m[3] = {{d(31), d(32), 515, 256}, {d(33), d(34), 256, 512},
                   {d(35), d(36), 512, 1024}};
  Layer fp3m[2] = {{d(37), d(38), 1536, 256}, {d(39), d(40), 256, 256}};
  Layer fp2m[2] = {{d(41), d(42), 576, 256}, {d(43), d(44), 256, 128}};
  Layer fp1m[2] = {{d(45), d(46), 134, 128}, {d(47), d(48), 128, 128}};
  Layer fem[2]  = {{d(49), d(50), 128, 256}, {d(51), d(52), 256, 128}};
  Layer attm[2] = {{d(53), d(54), 128, 64}, {d(55), d(56), 64, 128}};
  Layer conv1m  = {d(57), d(58), 128, 128};
  Layer conv2m  = {d(59), d(60), 128, 2};

  // ---- workspace bump allocator ----
  char* wsb = (char*)d_ws;
  size_t off = 0;
  auto allocF = [&](size_t n) { float* p = (float*)(wsb + off);
                                off += ((n * 4 + 255) / 256) * 256; return p; };
  auto allocI = [&](size_t n) { int* p = (int*)(wsb + off);
                                off += ((n * 4 + 255) / 256) * 256; return p; };

  float* pts      = allocF((size_t)B * N * 3);
  float* nxyz1    = allocF((size_t)B * S1 * 3);
  float* nxyz2    = allocF((size_t)B * S2 * 3);
  int*   fidx1    = allocI((size_t)B * S1);
  int*   fidx2    = allocI((size_t)B * S2);
  int*   gidx     = allocI((size_t)B * S1 * 128);
  float* l1pts    = allocF((size_t)B * S1 * 320);
  float* l2pts    = allocF((size_t)B * S2 * 512);
  float* l3pts    = allocF((size_t)B * 1024);
  float* l2f      = allocF((size_t)B * S2 * 256);
  float* l1f      = allocF((size_t)B * S1 * 128);
  float* xb       = allocF((size_t)B * N * 128);
  int*   idx3     = allocI((size_t)B * N * 3);
  float* w3       = allocF((size_t)B * N * 3);
  float* bufA     = allocF((size_t)B * S1 * 128 * 128);   // 67.1M floats
  float* bufB     = allocF((size_t)B * S1 * 128 * 128);

  auto NB = [](size_t n) { return dim3((unsigned)((n + 255) / 256)); };
  auto gemm = [&](const float* X, const Layer& L, float* Y, size_t rows, int mode) {
    size_t tiles = ((rows + 15) / 16) * (size_t)((L.cout + 63) / 64);
    unsigned blocks = (unsigned)((tiles + 7) / 8);     // 8 waves / 256-thr block
    k_gemm_wmma<<<blocks, 256, 0, stream>>>(X, L.W, L.b, Y, (int)rows,
                                            L.cin, L.cout, mode);
  };

  // ---- l0: transpose to (B,N,3); l0_points == l0_xyz == pts ----
  k_transpose<<<NB((size_t)B * 3 * N), 256, 0, stream>>>(xyz_bcn, pts, B, 3, N);

  // ---- SA1 ----
  k_fps<<<B, FPS_T, 0, stream>>>(pts, fidx1, N, S1);
  k_gather<<<NB((size_t)B * S1 * 3), 256, 0, stream>>>(pts, fidx1, nxyz1, B, N, S1, 3);

  auto run_scale = [&](const float* xyzP, const float* pointsP, int Cp, int Nn,
                       const float* centers, int S, float radius, int k,
                       const Layer* mlp, int nl,
                       float* outPts, int out_ld, int out_off) {
    float r2 = radius * radius;
    size_t waves = (size_t)B * S;
    k_query_ball<<<NB(waves * 32), 256, 0, stream>>>(xyzP, centers, gidx,
                                                     B, Nn, S, k, r2);
    size_t rows = (size_t)B * S * k;
    k_group_concat<<<NB(rows * (Cp + 3)), 256, 0, stream>>>(
        xyzP, pointsP, centers, gidx, bufA, B, Nn, S, k, Cp);
    const float* cur = bufA; float* nxt = bufB;
    for (int l = 0; l < nl; ++l) {
      gemm(cur, mlp[l], nxt, rows, 1);
      const float* t = cur; cur = nxt; nxt = (float*)t;
    }
    k_maxpool<<<NB((size_t)B * S * mlp[nl - 1].cout), 256, 0, stream>>>(
        cur, outPts, B * S, k, mlp[nl - 1].cout, out_ld, out_off);
  };

  run_scale(pts, pts, 3, N, nxyz1, S1, 0.1f,  32, sa1m[0], 3, l1pts, 320, 0);
  run_scale(pts, pts, 3, N, nxyz1, S1, 0.2f,  64, sa1m[1], 3, l1pts, 320, 64);
  run_scale(pts, pts, 3, N, nxyz1, S1, 0.4f, 128, sa1m[2], 3, l1pts, 320, 192);

  // ---- SA2 ----
  k_fps<<<B, FPS_T, 0, stream>>>(nxyz1, fidx2, S1, S2);
  k_gather<<<NB((size_t)B * S2 * 3), 256, 0, stream>>>(nxyz1, fidx2, nxyz2, B, S1, S2, 3);
  run_scale(nxyz1, l1pts, 320, S1, nxyz2, S2, 0.4f,  64, sa2m[0], 3, l2pts, 512, 0);
  run_scale(nxyz1, l1pts, 320, S1, nxyz2, S2, 0.8f, 128, sa2m[1], 3, l2pts, 512, 256);

  // ---- SA3 (group-all): g = [l2_xyz, l2_points] -> MLP -> max over 128 ----
  {
    size_t rows = (size_t)B * S2;
    k_concat2<<<NB(rows * 515), 256, 0, stream>>>(nxyz2, 3, l2pts, 512, bufA, rows);
    gemm(bufA, sa3m[0], bufB, rows, 1);
    gemm(bufB, sa3m[1], bufA, rows, 1);
    gemm(bufA, sa3m[2], bufB, rows, 1);
    k_maxpool<<<NB((size_t)B * 1024), 256, 0, stream>>>(bufB, l3pts, B, S2, 1024, 1024, 0);
  }

  // ---- FP3: broadcast l3 -> concat with l2_points -> MLP ----
  {
    size_t rows = (size_t)B * S2;
    k_nn_broadcast<<<NB(rows), 256, 0, stream>>>(idx3, w3, rows);
    k_interp_concat<<<NB(rows * 1536), 256, 0, stream>>>(
        l2pts, 512, nullptr, 0, l3pts, 1024, idx3, w3, bufA, B, S2, 1);
    gemm(bufA, fp3m[0], bufB, rows, 1);
    gemm(bufB, fp3m[1], l2f, rows, 1);
  }

  // ---- FP2: 3-NN l1<-l2 ----
  {
    size_t rows = (size_t)B * S1;
    k_nn3<<<NB(rows), 256, 0, stream>>>(nxyz1, nxyz2, idx3, w3, B, S1, S2);
    k_interp_concat<<<NB(rows * 576), 256, 0, stream>>>(
        l1pts, 320, nullptr, 0, l2f, 256, idx3, w3, bufA, B, S1, S2);
    gemm(bufA, fp2m[0], bufB, rows, 1);
    gemm(bufB, fp2m[1], l1f, rows, 1);
  }

  // ---- FP1: 3-NN l0<-l1; points1 = [l0_xyz, l0_points] = [pts, pts] ----
  size_t rows0 = (size_t)B * N;
  {
    k_nn3<<<NB(rows0), 256, 0, stream>>>(pts, nxyz1, idx3, w3, B, N, S1);
    k_interp_concat<<<NB(rows0 * 134), 256, 0, stream>>>(
        pts, 3, pts, 3, l1f, 128, idx3, w3, bufA, B, N, S1);
    gemm(bufA, fp1m[0], bufB, rows0, 1);
    gemm(bufB, fp1m[1], bufA, rows0, 1);   // l0_feat in bufA
  }

  // ---- head: fe -> attention -> conv1 -> conv2 ----
  gemm(bufA, fem[0], bufB, rows0, 1);
  gemm(bufB, fem[1], xb, rows0, 1);                 // x
  gemm(xb, attm[0], bufA, rows0, 1);
  gemm(bufA, attm[1], bufB, rows0, 2);              // a = sigmoid(...)
  k_mul<<<NB(rows0 * 128), 256, 0, stream>>>(xb, bufB, bufA, rows0 * 128);
  gemm(bufA, conv1m, bufB, rows0, 1);
  gemm(bufB, conv2m, (float*)d_out, rows0, 0);      // seg -> d_out[0 .. 262143]

  // ---- second output: transpose(l3_points,(0,2,1)) is flat-identical ----
  hipMemcpyAsync((float*)d_out + (size_t)B * N * 2, l3pts,
                 (size_t)B * 1024 * sizeof(float),
                 hipMemcpyDeviceToDevice, stream);
}